// ToneStack_24335284699686
// MI455X (gfx1250) — compile-verified
//
#include <hip/hip_runtime.h>

// ---------------------------------------------------------------------------
// ToneStack: 3 cascaded DF2T biquads over x[64][480000] f32.
// Fold the cascade into one 6-state LTI system; chunk-parallel scan breaks the
// time recurrence; per 256-sample group (16 blocks of 16):
//   U = V*X            (4x v_wmma_f32_16x16x4_f32)
//   s_{k+1}=A16 s_k+u_k (16-step shuffle scan, serial depth cut 16x)
//   Y = T16*X + W*S     (6x v_wmma_f32_16x16x4_f32)
// X staged global->LDS and Y drained LDS->global with CDNA5 async LDS DMA
// (global_load_async_to_lds_b128 / global_store_async_from_lds_b128).
// Roofline: 3 passes x 123MB = 369MB -> ~16us @ 23.3TB/s (2nd/3rd hit L2).
// ---------------------------------------------------------------------------

#define TLEN   480000
#define BROWS  64
#define CHUNK  1024
#define NCHUNK 469              // ceil(480000/1024); last chunk = 768 samples

// workspace layout (float offsets)
#define PT    0                 // T16 Toeplitz, 16x16 row-major
#define PWM   256               // W padded 16x16 (cols>=6 zero)
#define PVM   512               // V padded 16x16 (rows>=6 zero)
#define PA16  768               // A^16, 6x6
#define PMCH  804               // M = A^1024, 6x6
#define PVS   1024              // per-(row,chunk) zero-state end states, 6 ea
#define PSS   (PVS + BROWS*NCHUNK*6)   // per-(row,chunk) start states, 6 ea

typedef float v2f __attribute__((ext_vector_type(2)));
typedef float v8f __attribute__((ext_vector_type(8)));

static __device__ inline v8f wmma4(v2f a, v2f b, v8f c) {
  // D = A(16x4 f32) x B(4x16 f32) + C(16x16 f32)
  return __builtin_amdgcn_wmma_f32_16x16x4_f32(false, a, false, b,
                                               (short)0, c, false, false);
}

// Async LDS DMA: copy 32B/lane global -> LDS, wait for arrival.
// IOFFSET applies to both LDS and memory addresses (ISA 10.7/15.18), so one
// LDS-address VGPR serves both halves.  LDS operand = low 32 bits of the flat
// shared-pointer (flat LDS addressing truncates to wave-relative offset).
static __device__ inline void stage_lds_async(const float* gp, float* lp) {
  unsigned int lo = (unsigned int)(unsigned long long)lp;
  asm volatile(
      "global_load_async_to_lds_b128 %0, %1, off\n\t"
      "global_load_async_to_lds_b128 %0, %1, off offset:16\n\t"
      "s_wait_asynccnt 0x0"
      :: "v"(lo), "v"(gp)
      : "memory");
}

// Async LDS DMA: copy 32B/lane LDS -> global (drained by next group's wait /
// implicit S_WAIT_IDLE at S_ENDPGM).
static __device__ inline void drain_lds_async(float* gp, const float* lp) {
  unsigned int lo = (unsigned int)(unsigned long long)lp;
  asm volatile(
      "global_store_async_from_lds_b128 %0, %1, off\n\t"
      "global_store_async_from_lds_b128 %0, %1, off offset:16"
      :: "v"(gp), "v"(lo)
      : "memory");
}

// bilinear low-shelf, normalized coefficients (matches reference, f32 math)
static __device__ inline void shelf_coeffs(float fc, float gdb, float Q, float* o) {
  const float PIF = 3.14159265358979323846f;
  float A  = powf(10.0f, gdb / 40.0f);
  float w0 = 2.0f * PIF * fc / 48000.0f;
  float alpha = sinf(w0) / (2.0f * Q);
  float cw = cosf(w0);
  float sA = sqrtf(A);
  float b0 = A * ((A + 1.0f) - (A - 1.0f) * cw + 2.0f * sA * alpha);
  float b1 = 2.0f * A * ((A - 1.0f) - (A + 1.0f) * cw);
  float b2 = A * ((A + 1.0f) - (A - 1.0f) * cw - 2.0f * sA * alpha);
  float a0 = (A + 1.0f) + (A - 1.0f) * cw + 2.0f * sA * alpha;
  float a1 = -2.0f * ((A - 1.0f) + (A + 1.0f) * cw);
  float a2 = (A + 1.0f) + (A - 1.0f) * cw - 2.0f * sA * alpha;
  o[0] = b0 / a0; o[1] = b1 / a0; o[2] = b2 / a0; o[3] = a1 / a0; o[4] = a2 / a0;
}

// one combined-system step in double: s' , y  from (s, x)
static __device__ inline void dstep6(const float* c, const double* s, double x,
                                     double* sn, double* yo) {
  double y1 = (double)c[0]  * x  + s[0];
  sn[0]     = (double)c[1]  * x  - (double)c[3]  * y1 + s[1];
  sn[1]     = (double)c[2]  * x  - (double)c[4]  * y1;
  double y2 = (double)c[5]  * y1 + s[2];
  sn[2]     = (double)c[6]  * y1 - (double)c[8]  * y2 + s[3];
  sn[3]     = (double)c[7]  * y1 - (double)c[9]  * y2;
  double y3 = (double)c[10] * y2 + s[4];
  sn[4]     = (double)c[11] * y2 - (double)c[13] * y3 + s[5];
  sn[5]     = (double)c[12] * y2 - (double)c[14] * y3;
  *yo = y3;
}

// ---------------------------------------------------------------------------
__global__ void ts_setup(const float* lg, const float* mg, const float* mfc,
                         const float* mq, const float* hg, float* ws) {
  if (threadIdx.x != 0 || blockIdx.x != 0) return;

  float coef[15];
  shelf_coeffs(120.0f,  *lg, 0.707f, coef + 0);
  shelf_coeffs(*mfc,    *mg, *mq,    coef + 5);
  shelf_coeffs(4000.0f, *hg, 0.707f, coef + 10);

  // build A (6x6), b, c, d numerically from the step map
  double A[6][6], b6[6], c6[6], dd;
  double s[6], sn[6], y;
  for (int j = 0; j < 6; ++j) {
    for (int i = 0; i < 6; ++i) s[i] = (i == j) ? 1.0 : 0.0;
    dstep6(coef, s, 0.0, sn, &y);
    for (int i = 0; i < 6; ++i) A[i][j] = sn[i];
    c6[j] = y;
  }
  for (int i = 0; i < 6; ++i) s[i] = 0.0;
  dstep6(coef, s, 1.0, sn, &y);
  for (int i = 0; i < 6; ++i) b6[i] = sn[i];
  dd = y;

  // powers: W rows, V cols, impulse response h, A^16
  double Ai[6][6];
  for (int i = 0; i < 6; ++i)
    for (int j = 0; j < 6; ++j) Ai[i][j] = (i == j) ? 1.0 : 0.0;

  double h[17], Wd[16][6], Vd[6][16];
  for (int p = 0; p < 16; ++p) {
    for (int j = 0; j < 6; ++j) {              // W[p] = c^T A^p
      double acc = 0.0;
      for (int k = 0; k < 6; ++k) acc += c6[k] * Ai[k][j];
      Wd[p][j] = acc;
    }
    double t[6];
    for (int r = 0; r < 6; ++r) {              // t = A^p b
      double acc = 0.0;
      for (int k = 0; k < 6; ++k) acc += Ai[r][k] * b6[k];
      t[r] = acc;
    }
    {
      double acc = 0.0;
      for (int r = 0; r < 6; ++r) acc += c6[r] * t[r];
      h[p + 1] = acc;                          // h[m] = c^T A^{m-1} b
    }
    for (int r = 0; r < 6; ++r) Vd[r][15 - p] = t[r];   // V[:,j]=A^{15-j} b
    // Ai = A * Ai
    double tmp[6][6];
    for (int r = 0; r < 6; ++r)
      for (int cc = 0; cc < 6; ++cc) {
        double acc = 0.0;
        for (int k = 0; k < 6; ++k) acc += A[r][k] * Ai[k][cc];
        tmp[r][cc] = acc;
      }
    for (int r = 0; r < 6; ++r)
      for (int cc = 0; cc < 6; ++cc) Ai[r][cc] = tmp[r][cc];
  }
  // Ai == A^16.  M = (A^16)^64 = A^1024 by 6 squarings
  double Md[6][6];
  for (int r = 0; r < 6; ++r)
    for (int cc = 0; cc < 6; ++cc) Md[r][cc] = Ai[r][cc];
  for (int q = 0; q < 6; ++q) {
    double tmp[6][6];
    for (int r = 0; r < 6; ++r)
      for (int cc = 0; cc < 6; ++cc) {
        double acc = 0.0;
        for (int k = 0; k < 6; ++k) acc += Md[r][k] * Md[k][cc];
        tmp[r][cc] = acc;
      }
    for (int r = 0; r < 6; ++r)
      for (int cc = 0; cc < 6; ++cc) Md[r][cc] = tmp[r][cc];
  }

  // emit plan (f32)
  for (int i = 0; i < 16; ++i)
    for (int j = 0; j < 16; ++j) {
      ws[PT  + i*16 + j] = (j > i) ? 0.0f : ((j == i) ? (float)dd : (float)h[i - j]);
      ws[PWM + i*16 + j] = (j < 6) ? (float)Wd[i][j] : 0.0f;
      ws[PVM + i*16 + j] = (i < 6) ? (float)Vd[i][j] : 0.0f;
    }
  for (int i = 0; i < 6; ++i)
    for (int j = 0; j < 6; ++j) {
      ws[PA16 + i*6 + j] = (float)Ai[i][j];
      ws[PMCH + i*6 + j] = (float)Md[i][j];
    }
}

// ---------------------------------------------------------------------------
// Wave-per-chunk WMMA block processor.
//   EMIT=false: start from s=0, propagate chunk-end state -> vstates (phase 1)
//   EMIT=true : start from scanned state, emit filtered output    (phase 3)
template <bool EMIT>
__global__ __launch_bounds__(32) void ts_chunk(const float* __restrict__ x,
                                               float* __restrict__ ws,
                                               float* __restrict__ out) {
  __shared__ float ldsX[256];
  __shared__ float ldsU[256];
  __shared__ float ldsS[128];   // col-major: [blk*8 + stateDim], rows 6,7 zero
  __shared__ float ldsY[256];

  const int lane = threadIdx.x;
  const int lp   = lane & 15;
  const int hf   = lane >> 4;
  const int row  = blockIdx.y;
  const int ch   = blockIdx.x;
  const size_t idx = (size_t)row * NCHUNK + ch;

  const float* Vm  = ws + PVM;
  const float* A16 = ws + PA16;

  // A-matrix regs: VGPR v, lane-half h -> K = 4*kk + v + 2*h ; M = lp
  v2f Va[4];
#pragma unroll
  for (int kk = 0; kk < 4; ++kk)
    Va[kk] = *(const v2f*)(Vm + lp*16 + 4*kk + 2*hf);

  v2f Ta[4], Wa[2];
  if constexpr (EMIT) {
    const float* Tm = ws + PT;
    const float* Wm = ws + PWM;
#pragma unroll
    for (int kk = 0; kk < 4; ++kk)
      Ta[kk] = *(const v2f*)(Tm + lp*16 + 4*kk + 2*hf);
#pragma unroll
    for (int kk = 0; kk < 2; ++kk)
      Wa[kk] = *(const v2f*)(Wm + lp*16 + 4*kk + 2*hf);
  }

  const int rr = (lane < 6) ? lane : 5;
  float a16r[6];
#pragma unroll
  for (int j = 0; j < 6; ++j) a16r[j] = A16[rr*6 + j];

  float s = 0.0f;
  if constexpr (EMIT) {
    if (lane < 6) s = ws[PSS + idx*6 + lane];
    for (int i = lane; i < 128; i += 32) ldsS[i] = 0.0f;  // pad rows stay 0
  }

  const int base0 = row * TLEN + ch * CHUNK;
  int len = TLEN - ch * CHUNK; if (len > CHUNK) len = CHUNK;
  const int ngroups = len >> 8;                          // 4, last chunk 3

  for (int g = 0; g < ngroups; ++g) {
    const int base = base0 + g * 256;
    __syncthreads();
    // async DMA stage of 256 samples into LDS (32B per lane)
    stage_lds_async(x + base + lane*8, &ldsX[lane*8]);
    __syncthreads();

    // B-matrix regs for X: row K = 4*kk + v + 2*h, col N = lp
    v2f Xb4[4];
#pragma unroll
    for (int kk = 0; kk < 4; ++kk)
      Xb4[kk] = *(const v2f*)(&ldsX[lp*16 + 4*kk + 2*hf]);

    // U = Vpad * X  (per-block state injections, cols = blocks)
    v8f u = {0.f,0.f,0.f,0.f,0.f,0.f,0.f,0.f};
#pragma unroll
    for (int kk = 0; kk < 4; ++kk) u = wmma4(Va[kk], Xb4[kk], u);

#pragma unroll
    for (int r = 0; r < 8; ++r) ldsU[(r + 8*hf)*16 + lp] = u[r];
    __syncthreads();

    // 16-step block-state scan: s_{k+1} = A16 s_k + U[:,k]  (lanes 0..5 own s)
#pragma unroll 1
    for (int k = 0; k < 16; ++k) {
      if constexpr (EMIT) {
        if (lane < 6) ldsS[k*8 + lane] = s;
      }
      float uk = ldsU[rr*16 + k];
      float ns = uk;
#pragma unroll
      for (int j = 0; j < 6; ++j) ns = fmaf(a16r[j], __shfl(s, j, 32), ns);
      s = ns;                                   // after k=15: next group start
    }

    if constexpr (EMIT) {
      __syncthreads();
      // Y = T*X + W*S
      v8f y = {0.f,0.f,0.f,0.f,0.f,0.f,0.f,0.f};
#pragma unroll
      for (int kk = 0; kk < 4; ++kk) y = wmma4(Ta[kk], Xb4[kk], y);
      v2f Sb0 = *(const v2f*)(&ldsS[lp*8 + 2*hf]);        // K = 0..3
      v2f Sb1 = *(const v2f*)(&ldsS[lp*8 + 4 + 2*hf]);    // K = 4..7 (6,7 = 0)
      y = wmma4(Wa[0], Sb0, y);
      y = wmma4(Wa[1], Sb1, y);

      // transpose through LDS -> async DMA drain (coalesced 16B/lane x2)
#pragma unroll
      for (int r = 0; r < 8; ++r) ldsY[lp*16 + r + 8*hf] = y[r];
      __syncthreads();   // dscnt: ldsY visible before async store reads it
      drain_lds_async(out + base + lane*8, &ldsY[lane*8]);
      // next group's stage_lds_async waits asynccnt 0 before reusing ldsY;
      // final group covered by implicit S_WAIT_IDLE at S_ENDPGM.
    }
  }

  if constexpr (!EMIT) {
    if (lane < 6) ws[PVS + idx*6 + lane] = s;   // chunk-end state, zero init
  }
}

// ---------------------------------------------------------------------------
// Phase 2: per-row serial scan over chunks: s_{c+1} = M s_c + v_c.
__global__ void ts_phase2(float* __restrict__ ws) {
  int row = threadIdx.x;
  if (row >= BROWS) return;
  float M[36];
#pragma unroll
  for (int k = 0; k < 36; ++k) M[k] = ws[PMCH + k];

  const float* v = ws + PVS + (size_t)row * NCHUNK * 6;
  float*       o = ws + PSS + (size_t)row * NCHUNK * 6;
  float s[6] = {0.f, 0.f, 0.f, 0.f, 0.f, 0.f};
  for (int c = 0; c < NCHUNK; ++c) {
#pragma unroll
    for (int i = 0; i < 6; ++i) o[c*6 + i] = s[i];
    float ns[6];
#pragma unroll
    for (int i = 0; i < 6; ++i) {
      float acc = v[c*6 + i];
#pragma unroll
      for (int j = 0; j < 6; ++j) acc = fmaf(M[i*6 + j], s[j], acc);
      ns[i] = acc;
    }
#pragma unroll
    for (int i = 0; i < 6; ++i) s[i] = ns[i];
  }
}

// ---------------------------------------------------------------------------
extern "C" void kernel_launch(void* const* d_in, const int* in_sizes, int n_in,
                              void* d_out, int out_size, void* d_ws, size_t ws_size,
                              hipStream_t stream) {
  (void)in_sizes; (void)n_in; (void)out_size; (void)ws_size;
  const float* x   = (const float*)d_in[0];
  const float* lg  = (const float*)d_in[1];
  const float* mg  = (const float*)d_in[2];
  const float* mfc = (const float*)d_in[3];
  const float* mq  = (const float*)d_in[4];
  const float* hg  = (const float*)d_in[5];
  float* out = (float*)d_out;
  float* ws  = (float*)d_ws;

  ts_setup<<<1, 1, 0, stream>>>(lg, mg, mfc, mq, hg, ws);

  dim3 g3(NCHUNK, BROWS, 1);
  ts_chunk<false><<<g3, 32, 0, stream>>>(x, ws, out);   // phase 1: chunk states
  ts_phase2<<<1, BROWS, 0, stream>>>(ws);               // phase 2: chunk scan
  ts_chunk<true><<<g3, 32, 0, stream>>>(x, ws, out);    // phase 3: emit output
}